// GCNAutoencoder_22041772163208
// MI455X (gfx1250) — compile-verified
//
#include <hip/hip_runtime.h>

typedef __attribute__((ext_vector_type(2))) float v2f;
typedef __attribute__((ext_vector_type(8))) float v8f;

#define D_DIM 128
#define LDS_STRIDE 132          // padded transposed-W stride: conflict-free ds_load_b64
#define LN_EPS 1e-5f
#define NOISE_STD 0.1f

// ---------------------------------------------------------------- utilities
__global__ void zero_f32_kernel(float* __restrict__ p, long n) {
  long i = (long)blockIdx.x * blockDim.x + threadIdx.x;
  long stride = (long)gridDim.x * blockDim.x;
  for (; i < n; i += stride) p[i] = 0.0f;
}

// ---------------------------------------------------------------- degrees
__global__ void degree_kernel(const int* __restrict__ src, const int* __restrict__ dst,
                              float* __restrict__ deg_out, float* __restrict__ deg_in, int E) {
  int e = blockIdx.x * blockDim.x + threadIdx.x;
  if (e < E) {
    atomicAdd(&deg_out[src[e]], 1.0f);
    atomicAdd(&deg_in[dst[e]], 1.0f);
  }
}

__global__ void isqrt_kernel(float* __restrict__ deg_out, float* __restrict__ deg_in, int n) {
  int i = blockIdx.x * blockDim.x + threadIdx.x;
  if (i < n) {
    float a = deg_out[i];
    float b = deg_in[i];
    deg_out[i] = (a > 0.0f) ? rsqrtf(a) : 0.0f;   // deg>0 => deg>=1, max(deg,1)=deg
    deg_in[i]  = (b > 0.0f) ? rsqrtf(b) : 0.0f;
  }
}

// ------------------------------------------- mask + noise + D_out^-1/2 scale
// one block (128 threads) per node
__global__ void prep_kernel(const float* __restrict__ feat, const float* __restrict__ noise,
                            const unsigned char* __restrict__ mask, const float* __restrict__ token,
                            const float* __restrict__ dout_isqrt,
                            float* __restrict__ xs, float* __restrict__ out_feat,
                            float* __restrict__ out_mask) {
  int i = blockIdx.x;
  int d = threadIdx.x;
  long idx = (long)i * D_DIM + d;
  float f = feat[idx];
  bool m = mask[i] != 0;
  float x = (m ? token[d] : f) + noise[idx] * NOISE_STD;
  xs[idx] = x * dout_isqrt[i];
  out_feat[idx] = f;
  if (d == 0) out_mask[i] = m ? 1.0f : 0.0f;
}

// ---------------------------------------------------------------- fp32 WMMA GEMM
// out[nrows x 128] = X[nrows x 128] @ W[128 x 128]  (+bias)(+relu)
// MODE 0: out = acc              (pre-aggregation conv transform)
// MODE 1: out = relu(acc + bias) (decoder hidden)
// MODE 2: out = acc + bias       (decoder output)
// Block: 256 threads = 8 waves; wave w owns rows [blk*128 + 16w, +16).
// W is staged TRANSPOSED in LDS (sWt[col][k], stride 132) so each B fragment
// is one aligned ds_load_b64 with all 32 lanes on distinct banks.
template <int MODE>
__global__ __launch_bounds__(256)
void gemm_nk128_kernel(const float* __restrict__ X, const float* __restrict__ W,
                       const float* __restrict__ bias, float* __restrict__ out, int nrows) {
  __shared__ float sWt[D_DIM * LDS_STRIDE];         // 67.6 KB of 320 KB/WGP
  for (int i = threadIdx.x; i < D_DIM * D_DIM; i += blockDim.x) {
    int k   = i >> 7;                               // row of W  (K index)
    int col = i & 127;                              // col of W  (N index)
    sWt[col * LDS_STRIDE + k] = W[i];
  }
  __syncthreads();

  const int wave = threadIdx.x >> 5;
  const int lane = threadIdx.x & 31;
  const int m    = lane & 15;                       // A row / B,C column within tile
  const int lh   = lane >> 4;                       // lane-half selects K pair / M offset
  const int kk   = lh * 2;

  const int row0 = blockIdx.x * 128 + wave * 16;
  int grow = row0 + m;
  if (grow >= nrows) grow = nrows - 1;              // clamped read, store is guarded

  v8f acc[8] = {};

  for (int k0 = 0; k0 < D_DIM; k0 += 4) {
    // A fragment (16x4 f32): lanes 0-15 hold K=k0+0,1 ; lanes 16-31 hold K=k0+2,3
    v2f a = *(const v2f*)(X + (long)grow * D_DIM + k0 + kk);
#pragma unroll
    for (int t = 0; t < 8; ++t) {
      // B fragment (4x16 f32): lane = column; contiguous K pair -> one b64 LDS read
      v2f b = *(const v2f*)(&sWt[(t * 16 + m) * LDS_STRIDE + k0 + kk]);
      acc[t] = __builtin_amdgcn_wmma_f32_16x16x4_f32(
          /*neg_a=*/false, a, /*neg_b=*/false, b,
          /*c_mod=*/(short)0, acc[t], /*reuse_a=*/false, /*reuse_b=*/false);
    }
  }

  // D fragment: VGPR r -> M = r + 8*lh, N = m
#pragma unroll
  for (int t = 0; t < 8; ++t) {
#pragma unroll
    for (int r = 0; r < 8; ++r) {
      int row = row0 + r + 8 * lh;
      if (row < nrows) {
        int col = t * 16 + m;
        float v = acc[t][r];
        if (MODE >= 1) v += bias[col];
        if (MODE == 1) v = fmaxf(v, 0.0f);
        out[(long)row * D_DIM + col] = v;
      }
    }
  }
}

// ---------------------------------------------------------------- edge scatter (SpMM)
// one wave per edge; lane carries 4 consecutive f32 -> 128 per edge
__global__ __launch_bounds__(256)
void spmm_kernel(const float* __restrict__ h, const int* __restrict__ src,
                 const int* __restrict__ dst, float* __restrict__ agg, int E) {
  int e    = (int)((blockIdx.x * (long)blockDim.x + threadIdx.x) >> 5);
  int lane = threadIdx.x & 31;
  if (e >= E) return;
  int s = src[e];
  int t = dst[e];
  const float4 v = ((const float4*)(h + (long)s * D_DIM))[lane];
  float* ap = agg + (long)t * D_DIM + lane * 4;
  atomicAdd(ap + 0, v.x);
  atomicAdd(ap + 1, v.y);
  atomicAdd(ap + 2, v.z);
  atomicAdd(ap + 3, v.w);
}

// ------------------------------- agg*Din^-1/2 + b  -> LayerNorm -> (relu*Dout | id)
// one block (128 threads) per node
// MODE 0: out = relu(LN(h)) * dout_isqrt   (feeds next conv, pre-scaled)
// MODE 1: out = LN(h)                       (z)
template <int MODE>
__global__ void postconv_kernel(const float* __restrict__ agg, const float* __restrict__ din_isqrt,
                                const float* __restrict__ bias, const float* __restrict__ g,
                                const float* __restrict__ beta, const float* __restrict__ dout_isqrt,
                                float* __restrict__ out) {
  __shared__ float red[D_DIM];
  int i = blockIdx.x;
  int d = threadIdx.x;
  float v = agg[(long)i * D_DIM + d] * din_isqrt[i] + bias[d];

  red[d] = v;
  __syncthreads();
  for (int s = 64; s > 0; s >>= 1) {
    if (d < s) red[d] += red[d + s];
    __syncthreads();
  }
  float mu = red[0] * (1.0f / D_DIM);
  __syncthreads();

  float c = v - mu;
  red[d] = c * c;
  __syncthreads();
  for (int s = 64; s > 0; s >>= 1) {
    if (d < s) red[d] += red[d + s];
    __syncthreads();
  }
  float var = red[0] * (1.0f / D_DIM);

  float ln = c * rsqrtf(var + LN_EPS) * g[d] + beta[d];
  if (MODE == 0)
    out[(long)i * D_DIM + d] = fmaxf(ln, 0.0f) * dout_isqrt[i];
  else
    out[(long)i * D_DIM + d] = ln;
}

// ---------------------------------------------------------------- classifier head
__global__ void logits_kernel(const float* __restrict__ z, const float* __restrict__ W,
                              const float* __restrict__ b, float* __restrict__ out, int n) {
  int t = blockIdx.x * blockDim.x + threadIdx.x;
  int node = t >> 4;
  int c    = t & 15;
  if (node >= n || c >= 10) return;
  float acc = b[c];
  const float* zr = z + (long)node * D_DIM;
#pragma unroll 4
  for (int d = 0; d < D_DIM; ++d) acc += zr[d] * W[d * 10 + c];
  out[node * 10 + c] = acc;
}

// ---------------------------------------------------------------- launch
extern "C" void kernel_launch(void* const* d_in, const int* in_sizes, int n_in,
                              void* d_out, int out_size, void* d_ws, size_t ws_size,
                              hipStream_t stream) {
  const float* feat   = (const float*)d_in[0];
  const float* noise  = (const float*)d_in[1];
  const unsigned char* nmask = (const unsigned char*)d_in[2];   // jnp bool -> 1 byte
  const int*   src    = (const int*)d_in[3];
  const int*   dst    = (const int*)d_in[4];
  const float* token  = (const float*)d_in[5];
  const float* W0     = (const float*)d_in[6];
  const float* b0     = (const float*)d_in[7];
  const float* ln0_g  = (const float*)d_in[8];
  const float* ln0_b  = (const float*)d_in[9];
  const float* W1     = (const float*)d_in[10];
  const float* b1     = (const float*)d_in[11];
  const float* ln1_g  = (const float*)d_in[12];
  const float* ln1_b  = (const float*)d_in[13];
  const float* dec_W1 = (const float*)d_in[14];
  const float* dec_b1 = (const float*)d_in[15];
  const float* dec_W2 = (const float*)d_in[16];
  const float* dec_b2 = (const float*)d_in[17];
  const float* cls_W  = (const float*)d_in[18];
  const float* cls_b  = (const float*)d_in[19];

  const int N = in_sizes[2];      // 50000
  const int E = in_sizes[3];      // 800000
  const long ND = (long)N * D_DIM;

  // output layout: x_recon[N*128] | feat[N*128] | mask[N] | z[N*128] | logits[N*10]
  float* out_recon  = (float*)d_out;
  float* out_feat   = out_recon + ND;
  float* out_mask   = out_feat + ND;
  float* out_z      = out_mask + N;
  float* out_logits = out_z + ND;

  // workspace layout
  float* bufA = (float*)d_ws;          // xs / xs1 / decoder hidden
  float* bufB = bufA + ND;             // pre-aggregation conv transform
  float* bufC = bufB + ND;             // aggregation accumulator
  float* dout_isqrt = bufC + ND;       // N
  float* din_isqrt  = dout_isqrt + N;  // N

  const int nblk_gemm = (N + 127) / 128;
  const int nblk_edge = (E + 7) / 8;                // 1 wave per edge, 8 per block

  // degrees -> isqrt
  zero_f32_kernel<<<1024, 256, 0, stream>>>(dout_isqrt, 2L * N);
  degree_kernel<<<(E + 255) / 256, 256, 0, stream>>>(src, dst, dout_isqrt, din_isqrt, E);
  isqrt_kernel<<<(N + 255) / 256, 256, 0, stream>>>(dout_isqrt, din_isqrt, N);

  // mask + noise + pre-scale; also emit feat/mask outputs
  prep_kernel<<<N, D_DIM, 0, stream>>>(feat, noise, nmask, token, dout_isqrt,
                                       bufA, out_feat, out_mask);

  // conv0
  gemm_nk128_kernel<0><<<nblk_gemm, 256, 0, stream>>>(bufA, W0, nullptr, bufB, N);
  zero_f32_kernel<<<4096, 256, 0, stream>>>(bufC, ND);
  spmm_kernel<<<nblk_edge, 256, 0, stream>>>(bufB, src, dst, bufC, E);
  postconv_kernel<0><<<N, D_DIM, 0, stream>>>(bufC, din_isqrt, b0, ln0_g, ln0_b,
                                              dout_isqrt, bufA);

  // conv1 -> z
  gemm_nk128_kernel<0><<<nblk_gemm, 256, 0, stream>>>(bufA, W1, nullptr, bufB, N);
  zero_f32_kernel<<<4096, 256, 0, stream>>>(bufC, ND);
  spmm_kernel<<<nblk_edge, 256, 0, stream>>>(bufB, src, dst, bufC, E);
  postconv_kernel<1><<<N, D_DIM, 0, stream>>>(bufC, din_isqrt, b1, ln1_g, ln1_b,
                                              dout_isqrt, out_z);

  // decoder
  gemm_nk128_kernel<1><<<nblk_gemm, 256, 0, stream>>>(out_z, dec_W1, dec_b1, bufA, N);
  gemm_nk128_kernel<2><<<nblk_gemm, 256, 0, stream>>>(bufA, dec_W2, dec_b2, out_recon, N);

  // classifier
  logits_kernel<<<((long)N * 16 + 255) / 256, 256, 0, stream>>>(out_z, cls_W, cls_b,
                                                                out_logits, N);
}